// ResidualGCNLayer_20693152432667
// MI455X (gfx1250) — compile-verified
//
#include <hip/hip_runtime.h>

#define D 128
#define BN_EPS 1e-5f

typedef float v2f __attribute__((ext_vector_type(2)));
typedef float v8f __attribute__((ext_vector_type(8)));

// ---------------- utility: zero fill ----------------
__global__ void zero_f32(float* __restrict__ p, long long n) {
    long long i = (long long)blockIdx.x * blockDim.x + threadIdx.x;
    long long stride = (long long)gridDim.x * blockDim.x;
    for (; i < n; i += stride) p[i] = 0.0f;
}

// ---------------- degree accumulation (in-degree via dst) ----------------
__global__ void degree_kernel(const long long* __restrict__ dst,
                              float* __restrict__ deg, int E) {
    int e = blockIdx.x * blockDim.x + threadIdx.x;
    if (e < E) atomicAdd(&deg[dst[e]], 1.0f);
}

// dinv[n] = rsqrt(indeg + 1)  (self-loop guarantees deg >= 1)
__global__ void dinv_kernel(const float* __restrict__ deg,
                            float* __restrict__ dinv, int n) {
    int i = blockIdx.x * blockDim.x + threadIdx.x;
    if (i < n) dinv[i] = rsqrtf(deg[i] + 1.0f);
}

// ---------------- H = X * W  via V_WMMA_F32_16X16X4_F32 ----------------
// Block: 256 threads = 8 waves. Block handles 16 nodes; wave w handles
// output-feature columns [16w, 16w+16). K loop over D=128 in steps of 4.
//
// A frag (16x4 f32, 2 VGPRs/lane):  lane l: m = l&15, kb = (l>>4)*2
//   a[0] = X[m][k+kb], a[1] = X[m][k+kb+1]
// B frag (4x16 f32, 2 VGPRs/lane):  lane l: n = l&15, kb = (l>>4)*2
//   b[0] = W[k+kb][n], b[1] = W[k+kb+1][n]
// D frag (16x16 f32, 8 VGPRs/lane): lane l, vgpr v -> row v + (l>=16 ? 8:0), col l&15
__global__ void __launch_bounds__(256)
gemm_xw_wmma(const float* __restrict__ X, const float* __restrict__ Wm,
             float* __restrict__ H, int N) {
    const int lane = threadIdx.x & 31;
    const int wave = threadIdx.x >> 5;
    const int m0   = blockIdx.x << 4;    // node tile base
    const int n0   = wave << 4;          // feature tile base
    const int mi   = lane & 15;
    const int kb   = (lane >> 4) << 1;   // 0 or 2

    int row = m0 + mi;
    if (row >= N) row = N - 1;           // clamp (keeps EXEC all-ones)
    const float* __restrict__ xrow = X  + (long long)row * D;
    const float* __restrict__ wcol = Wm + n0 + mi;

    v8f acc = {};
#pragma unroll
    for (int k = 0; k < D; k += 4) {
        v2f a, b;
        a[0] = xrow[k + kb];
        a[1] = xrow[k + kb + 1];
        b[0] = wcol[(long long)(k + kb) * D];
        b[1] = wcol[(long long)(k + kb + 1) * D];
        acc = __builtin_amdgcn_wmma_f32_16x16x4_f32(
            /*neg_a=*/false, a, /*neg_b=*/false, b,
            /*c_mod=*/(short)0, acc, /*reuse_a=*/false, /*reuse_b=*/false);
    }

    const int col = n0 + mi;
    const int mh  = (lane >> 4) << 3;    // 0 or 8
#pragma unroll
    for (int v = 0; v < 8; ++v) {
        int r = m0 + mh + v;
        if (r < N) H[(long long)r * D + col] = acc[v];
    }
}

// ---------------- per-edge normalized scatter-add ----------------
// One wave (32 lanes) per edge; each lane moves 4 consecutive features
// (float4 gather from h[src], 4 f32 atomic adds into out[dst]).
// h and out both fit in the 192 MB L2, so these atomics stay on-chip.
__global__ void __launch_bounds__(256)
edge_scatter(const float* __restrict__ H,
             const long long* __restrict__ src,
             const long long* __restrict__ dst,
             const float* __restrict__ dinv,
             float* __restrict__ out, int E) {
    long long gid = (long long)blockIdx.x * blockDim.x + threadIdx.x;
    int e    = (int)(gid >> 5);
    int lane = (int)(gid & 31);
    if (e >= E) return;
    long long s = src[e];
    long long d = dst[e];
    float w = dinv[s] * dinv[d];
    const float4 hv = *(const float4*)(H + s * D + lane * 4);
    float* op = out + d * D + lane * 4;
    atomicAdd(op + 0, hv.x * w);
    atomicAdd(op + 1, hv.y * w);
    atomicAdd(op + 2, hv.z * w);
    atomicAdd(op + 3, hv.w * w);
}

// ---------------- finalize: self-loop + bias + BN + ReLU + residual ----------
__global__ void __launch_bounds__(256)
finalize_kernel(const float* __restrict__ X, const float* __restrict__ H,
                const float* __restrict__ dinv,
                const float* __restrict__ bias,
                const float* __restrict__ gamma,
                const float* __restrict__ beta,
                const float* __restrict__ rmean,
                const float* __restrict__ rvar,
                float* __restrict__ out, int N) {
    long long i = (long long)blockIdx.x * blockDim.x + threadIdx.x;
    long long total = (long long)N * D;
    if (i >= total) return;
    int f = (int)(i & (D - 1));
    long long n = i >> 7;                 // D == 128
    float di  = dinv[n];
    float val = out[i] + H[i] * di * di + bias[f];   // add self-loop + bias
    float bn  = (val - rmean[f]) * rsqrtf(rvar[f] + BN_EPS) * gamma[f] + beta[f];
    out[i] = fmaxf(bn, 0.0f) + X[i];
}

// ---------------- launch ----------------
extern "C" void kernel_launch(void* const* d_in, const int* in_sizes, int n_in,
                              void* d_out, int out_size, void* d_ws, size_t ws_size,
                              hipStream_t stream) {
    const float*     x     = (const float*)d_in[0];
    const long long* ei    = (const long long*)d_in[1];   // [2, E] int64
    const float*     Wm    = (const float*)d_in[2];
    const float*     bias  = (const float*)d_in[3];
    const float*     gamma = (const float*)d_in[4];
    const float*     beta  = (const float*)d_in[5];
    const float*     rmean = (const float*)d_in[6];
    const float*     rvar  = (const float*)d_in[7];

    const int N = in_sizes[0] / D;
    const int E = in_sizes[1] / 2;
    const long long* src = ei;
    const long long* dst = ei + E;

    float* out  = (float*)d_out;
    float* H    = (float*)d_ws;                  // N*D floats
    float* deg  = H + (long long)N * D;          // N floats
    float* dinv = deg + N;                       // N floats

    // zero accumulator (d_out) and degree buffer every call (deterministic)
    zero_f32<<<2048, 256, 0, stream>>>(out, (long long)N * D);
    zero_f32<<<256, 256, 0, stream>>>(deg, N);

    degree_kernel<<<(E + 255) / 256, 256, 0, stream>>>(dst, deg, E);
    dinv_kernel<<<(N + 255) / 256, 256, 0, stream>>>(deg, dinv, N);

    gemm_xw_wmma<<<(N + 15) / 16, 256, 0, stream>>>(x, Wm, H, N);

    long long sthreads = (long long)E * 32;
    edge_scatter<<<(int)((sthreads + 255) / 256), 256, 0, stream>>>(
        H, src, dst, dinv, out, E);

    finalize_kernel<<<(int)(((long long)N * D + 255) / 256), 256, 0, stream>>>(
        x, H, dinv, bias, gamma, beta, rmean, rvar, out, N);
}